// MultiHeadAttention_47553877901359
// MI455X (gfx1250) — compile-verified
//
#include <hip/hip_runtime.h>

// MI455X / gfx1250, wave32. All matrix math via V_WMMA_F32_16X16X4_F32
// (exact f32 -> matches the f32 reference numerically while using the WMMA pipe).
//
// Workspace requirement: 3 * 32768 * 1024 * 4 bytes = 402,653,184 bytes.

typedef float v2f __attribute__((ext_vector_type(2)));
typedef float v8f __attribute__((ext_vector_type(8)));

#define M_TOT 32768   // B*S
#define N_TOT 1024    // F
#define K_TOT 1024    // F

// ---------------------------------------------------------------------------
// GEMM: C[M,N] = A[M,K] @ W[K,N] + bias[N]
// Block: 256 threads (8 waves as 4(M) x 2(N)); WG tile 128x128; wave tile 32x64.
// K staged 16 at a time in LDS, double buffered (single barrier scheme).
// A staged row-major (As[m][k], stride 20); B staged TRANSPOSED (Bt[n][k],
// stride 20) so every WMMA B-fragment is one aligned 8B ds_load_b64 into an
// even VGPR pair -> no v_mov repacking in the inner loop.
// ---------------------------------------------------------------------------
#define AS_STRIDE 20
#define BT_STRIDE 20

__global__ __launch_bounds__(256)
void gemm_wmma_f32(const float* __restrict__ A, const float* __restrict__ W,
                   const float* __restrict__ bias, float* __restrict__ C) {
  __shared__ float As[2][128 * AS_STRIDE];   // 20480 B
  __shared__ float Bt[2][128 * BT_STRIDE];   // 20480 B

  const int tid   = threadIdx.x;
  const int lane  = tid & 31;
  const int wave  = tid >> 5;
  const int waveM = wave & 3;     // 0..3
  const int waveN = wave >> 2;    // 0..1
  const int blockM = blockIdx.y * 128;
  const int blockN = blockIdx.x * 128;

  v8f acc[2][4];
#pragma unroll
  for (int mt = 0; mt < 2; ++mt)
#pragma unroll
    for (int nt = 0; nt < 4; ++nt)
#pragma unroll
      for (int i = 0; i < 8; ++i) acc[mt][nt][i] = 0.0f;

  // A stage mapping: 128 rows x 16 cols, float4 per thread (x2).
  const int a_row = tid >> 2;               // 0..63 (and +64)
  const int a_col = (tid & 3) << 2;         // 0,4,8,12
  // B stage mapping (transposed): each thread owns one n column, 8 k values.
  const int bt_n  = tid & 127;              // 0..127
  const int bt_k0 = (tid >> 7) << 3;        // 0 or 8

  const float* Ap = A + (size_t)blockM * K_TOT;
  const float* Wp = W + blockN;

  const int rq   = lane & 15;               // fragment row/col within 16
  const int koff = (lane >> 4) << 1;        // 0 or 2 (K sub-offset per lane half)

  float4 ra0, ra1;
  float  rb[8];

  // prologue: stage k0 = 0 into buffer 0
  ra0 = *(const float4*)(Ap + (size_t)a_row * K_TOT + a_col);
  ra1 = *(const float4*)(Ap + (size_t)(a_row + 64) * K_TOT + a_col);
#pragma unroll
  for (int j = 0; j < 8; ++j)
    rb[j] = Wp[(size_t)(bt_k0 + j) * N_TOT + bt_n];
  *(float4*)(&As[0][a_row * AS_STRIDE + a_col])        = ra0;
  *(float4*)(&As[0][(a_row + 64) * AS_STRIDE + a_col]) = ra1;
  {
    float4 s0 = make_float4(rb[0], rb[1], rb[2], rb[3]);
    float4 s1 = make_float4(rb[4], rb[5], rb[6], rb[7]);
    *(float4*)(&Bt[0][bt_n * BT_STRIDE + bt_k0])     = s0;
    *(float4*)(&Bt[0][bt_n * BT_STRIDE + bt_k0 + 4]) = s1;
  }
  __syncthreads();

  int buf = 0;
  for (int k0 = 16; k0 <= K_TOT; k0 += 16) {
    const bool more = (k0 < K_TOT);
    if (more) {  // prefetch next K-slab into registers
      ra0 = *(const float4*)(Ap + (size_t)a_row * K_TOT + k0 + a_col);
      ra1 = *(const float4*)(Ap + (size_t)(a_row + 64) * K_TOT + k0 + a_col);
#pragma unroll
      for (int j = 0; j < 8; ++j)
        rb[j] = Wp[(size_t)(k0 + bt_k0 + j) * N_TOT + bt_n];
    }
    // compute current slab from LDS
#pragma unroll
    for (int kk = 0; kk < 16; kk += 4) {
      v2f af[2], bf[4];
      const int kc = kk + koff;
#pragma unroll
      for (int mt = 0; mt < 2; ++mt)
        af[mt] = *(const v2f*)(&As[buf][(waveM * 32 + mt * 16 + rq) * AS_STRIDE + kc]);
#pragma unroll
      for (int nt = 0; nt < 4; ++nt)
        bf[nt] = *(const v2f*)(&Bt[buf][(waveN * 64 + nt * 16 + rq) * BT_STRIDE + kc]);
#pragma unroll
      for (int mt = 0; mt < 2; ++mt)
#pragma unroll
        for (int nt = 0; nt < 4; ++nt)
          acc[mt][nt] = __builtin_amdgcn_wmma_f32_16x16x4_f32(
              false, af[mt], false, bf[nt], (short)0, acc[mt][nt], false, false);
    }
    if (more) {  // commit prefetched slab to the other buffer
      const int nb = buf ^ 1;
      *(float4*)(&As[nb][a_row * AS_STRIDE + a_col])        = ra0;
      *(float4*)(&As[nb][(a_row + 64) * AS_STRIDE + a_col]) = ra1;
      float4 s0 = make_float4(rb[0], rb[1], rb[2], rb[3]);
      float4 s1 = make_float4(rb[4], rb[5], rb[6], rb[7]);
      *(float4*)(&Bt[nb][bt_n * BT_STRIDE + bt_k0])     = s0;
      *(float4*)(&Bt[nb][bt_n * BT_STRIDE + bt_k0 + 4]) = s1;
      __syncthreads();
      buf = nb;
    }
  }

  // epilogue: C/D layout -> M = r + 8*(lane/16), N across lanes 0..15
  const int mhalf = (lane >> 4) << 3;
#pragma unroll
  for (int mt = 0; mt < 2; ++mt) {
    const int rowBase = blockM + waveM * 32 + mt * 16 + mhalf;
#pragma unroll
    for (int nt = 0; nt < 4; ++nt) {
      const int col = blockN + waveN * 64 + nt * 16 + rq;
      const float bv = bias[col];
#pragma unroll
      for (int r = 0; r < 8; ++r)
        C[(size_t)(rowBase + r) * N_TOT + col] = acc[mt][nt][r] + bv;
    }
  }
}

// ---------------------------------------------------------------------------
// Attention: per position p = (b,s): E = Q(16x64) Kt / 8, softmax over key-head,
// O = A(16x16) V(16x64).  One wave per position, 4 positions per 128-thread block.
// q/k staged row-major (stride 68); V staged TRANSPOSED (vt[d][g], stride 20)
// so its WMMA B-fragments are single 8B ds loads. attn bounced via LDS (C->A).
// Writes output in place over the q buffer (each position owned by one wave).
// ---------------------------------------------------------------------------
#define QS 68
#define VTS 20
#define ATS 20

__global__ __launch_bounds__(128)
void attn_kernel(const float* __restrict__ q, const float* __restrict__ k,
                 const float* __restrict__ v, float* __restrict__ out) {
  __shared__ float qs[4][16 * QS];
  __shared__ float ks[4][16 * QS];
  __shared__ float vt[4][64 * VTS];
  __shared__ float as_[4][16 * ATS];

  const int tid = threadIdx.x;
  const int base_p = blockIdx.x * 4;

  // cooperative staging of q,k,v for 4 positions (contiguous 1024-f32 rows)
  for (int j = 0; j < 4; ++j) {
    const size_t off = (size_t)(base_p + j) * 1024;
    for (int f = tid; f < 256; f += 128) {
      const int head = f >> 4;
      const int d4 = (f & 15) << 2;
      const float4 tq = *(const float4*)(q + off + (head << 6) + d4);
      const float4 tk = *(const float4*)(k + off + (head << 6) + d4);
      const float4 tv = *(const float4*)(v + off + (head << 6) + d4);
      float* dq = &qs[j][head * QS + d4];
      dq[0] = tq.x; dq[1] = tq.y; dq[2] = tq.z; dq[3] = tq.w;
      float* dk = &ks[j][head * QS + d4];
      dk[0] = tk.x; dk[1] = tk.y; dk[2] = tk.z; dk[3] = tk.w;
      vt[j][(d4 + 0) * VTS + head] = tv.x;   // transposed: vt[d][head]
      vt[j][(d4 + 1) * VTS + head] = tv.y;
      vt[j][(d4 + 2) * VTS + head] = tv.z;
      vt[j][(d4 + 3) * VTS + head] = tv.w;
    }
  }
  __syncthreads();

  const int w = tid >> 5;
  const int lane = tid & 31;
  const int rq = lane & 15;
  const int koff = (lane >> 4) << 1;
  const int mhalf = (lane >> 4) << 3;
  const float* Qd = &qs[w][0];
  const float* Kd = &ks[w][0];
  const float* Vt = &vt[w][0];
  float* Ad = &as_[w][0];

  // energy = Q @ K^T over D=64 (16 chained WMMAs)
  v8f e;
#pragma unroll
  for (int i = 0; i < 8; ++i) e[i] = 0.0f;
#pragma unroll
  for (int kk = 0; kk < 64; kk += 4) {
    const v2f a = *(const v2f*)(Qd + rq * QS + kk + koff);   // A[m=rq][k]
    const v2f b = *(const v2f*)(Kd + rq * QS + kk + koff);   // B[k][n=rq] = K[n][k]
    e = __builtin_amdgcn_wmma_f32_16x16x4_f32(false, a, false, b, (short)0, e,
                                              false, false);
  }

  // softmax over N (16 lanes within each half-wave), scale 1/sqrt(64)
#pragma unroll
  for (int r = 0; r < 8; ++r) {
    float x = e[r] * 0.125f;
    float m = x;
#pragma unroll
    for (int s = 1; s < 16; s <<= 1) m = fmaxf(m, __shfl_xor(m, s, 32));
    const float ex = __expf(x - m);
    float sum = ex;
#pragma unroll
    for (int s = 1; s < 16; s <<= 1) sum += __shfl_xor(sum, s, 32);
    e[r] = ex / sum;
  }

  // bounce attn through LDS to re-layout C -> A
#pragma unroll
  for (int r = 0; r < 8; ++r) Ad[(r + mhalf) * ATS + rq] = e[r];

  const size_t off = (size_t)(base_p + w) * 1024;
#pragma unroll
  for (int nt = 0; nt < 4; ++nt) {
    v8f o;
#pragma unroll
    for (int i = 0; i < 8; ++i) o[i] = 0.0f;
#pragma unroll
    for (int kk = 0; kk < 16; kk += 4) {
      const v2f a = *(const v2f*)(Ad + rq * ATS + kk + koff);
      const v2f b = *(const v2f*)(Vt + (nt * 16 + rq) * VTS + kk + koff);
      o = __builtin_amdgcn_wmma_f32_16x16x4_f32(false, a, false, b, (short)0, o,
                                                false, false);
    }
#pragma unroll
    for (int r = 0; r < 8; ++r)
      out[off + (size_t)(r + mhalf) * 64 + nt * 16 + rq] = o[r];
  }
}

// ---------------------------------------------------------------------------
// LayerNorm over last dim (1024), affine, eps=1e-5. One 256-thread block per row.
// ---------------------------------------------------------------------------
__global__ __launch_bounds__(256)
void layernorm_kernel(const float* __restrict__ x, const float* __restrict__ gamma,
                      const float* __restrict__ beta, float* __restrict__ out) {
  __shared__ float red[16];
  const int tid = threadIdx.x;
  const size_t row = blockIdx.x;
  const float* xr = x + row * 1024;

  const float4 vx = *(const float4*)(xr + tid * 4);
  float s  = vx.x + vx.y + vx.z + vx.w;
  float s2 = vx.x * vx.x + vx.y * vx.y + vx.z * vx.z + vx.w * vx.w;
#pragma unroll
  for (int sh = 1; sh < 32; sh <<= 1) {
    s  += __shfl_xor(s, sh, 32);
    s2 += __shfl_xor(s2, sh, 32);
  }
  if ((tid & 31) == 0) {
    red[tid >> 5] = s;
    red[8 + (tid >> 5)] = s2;
  }
  __syncthreads();
  float ts = 0.0f, ts2 = 0.0f;
#pragma unroll
  for (int i = 0; i < 8; ++i) { ts += red[i]; ts2 += red[8 + i]; }
  const float mu  = ts * (1.0f / 1024.0f);
  const float var = ts2 * (1.0f / 1024.0f) - mu * mu;
  const float rstd = rsqrtf(var + 1e-5f);

  const float4 g = *(const float4*)(gamma + tid * 4);
  const float4 bb = *(const float4*)(beta + tid * 4);
  float4 o;
  o.x = (vx.x - mu) * rstd * g.x + bb.x;
  o.y = (vx.y - mu) * rstd * g.y + bb.y;
  o.z = (vx.z - mu) * rstd * g.z + bb.z;
  o.w = (vx.w - mu) * rstd * g.w + bb.w;
  *(float4*)(out + row * 1024 + tid * 4) = o;
}

// ---------------------------------------------------------------------------
extern "C" void kernel_launch(void* const* d_in, const int* in_sizes, int n_in,
                              void* d_out, int out_size, void* d_ws, size_t ws_size,
                              hipStream_t stream) {
  (void)in_sizes; (void)n_in; (void)out_size; (void)ws_size;
  const float* x     = (const float*)d_in[0];
  const float* Wq    = (const float*)d_in[1];
  const float* bq    = (const float*)d_in[2];
  const float* Wk    = (const float*)d_in[3];
  const float* bk    = (const float*)d_in[4];
  const float* Wv    = (const float*)d_in[5];
  const float* bv    = (const float*)d_in[6];
  const float* Wo    = (const float*)d_in[7];
  const float* bo    = (const float*)d_in[8];
  const float* gamma = (const float*)d_in[9];
  const float* beta  = (const float*)d_in[10];

  float* qb = (float*)d_ws;                         // 32768*1024 f32
  float* kb = qb + (size_t)M_TOT * N_TOT;           // 32768*1024 f32
  float* vb = kb + (size_t)M_TOT * N_TOT;           // 32768*1024 f32

  const dim3 gemmGrid(N_TOT / 128, M_TOT / 128);    // (8, 256)
  gemm_wmma_f32<<<gemmGrid, 256, 0, stream>>>(x, Wq, bq, qb);
  gemm_wmma_f32<<<gemmGrid, 256, 0, stream>>>(x, Wk, bk, kb);
  gemm_wmma_f32<<<gemmGrid, 256, 0, stream>>>(x, Wv, bv, vb);

  attn_kernel<<<M_TOT / 4, 128, 0, stream>>>(qb, kb, vb, qb);  // in-place over qb

  gemm_wmma_f32<<<gemmGrid, 256, 0, stream>>>(qb, Wo, bo, kb);

  layernorm_kernel<<<M_TOT, 256, 0, stream>>>(kb, gamma, beta, (float*)d_out);
}